// e_ASPP_7249904796027
// MI455X (gfx1250) — compile-verified
//
#include <hip/hip_runtime.h>

#define Bc   8
#define CIN  256
#define Cr   64
#define Dn   4
#define Hh   128
#define Ww   128
#define HW   16384
#define COUT 256
#define NPIX (Bc*HW)   /* 131072 pixels per channel for BN stats */

typedef float v2f __attribute__((ext_vector_type(2)));
typedef float v8f __attribute__((ext_vector_type(8)));

__device__ __forceinline__ v8f wmma_f32(v2f a, v2f b, v8f c) {
    // V_WMMA_F32_16X16X4_F32 : D = A(16x4) x B(4x16) + C(16x16), fp32
    return __builtin_amdgcn_wmma_f32_16x16x4_f32(false, a, false, b, (short)0, c,
                                                 false, false);
}

__device__ __forceinline__ float red16(float v) {
    #pragma unroll
    for (int off = 8; off; off >>= 1) v += __shfl_xor(v, off, 32);
    return v;
}
__device__ __forceinline__ float red32(float v) {
    #pragma unroll
    for (int off = 16; off; off >>= 1) v += __shfl_xor(v, off, 32);
    return v;
}

// ---------------------------------------------------------------- init stats
__global__ __launch_bounds__(256) void k_init(float* p) {
    p[blockIdx.x * 256 + threadIdx.x] = 0.f;
}

// ------------------------------------------------ GEMM1: y_pre = cr_w @ x + b
// grid (128 pixel-blocks, 8 batch), block 256 = 8 waves.
// Each wave: Mtile(16 of C=64) x N=64 px; block covers C=64 x 128 px.
__global__ __launch_bounds__(256)
void k_gemm1(const float* __restrict__ x, const float* __restrict__ cr_w,
             const float* __restrict__ cr_b, float* __restrict__ y_pre,
             float* __restrict__ stats) {
    __shared__ float xs[32 * 128];
    __shared__ float as[64 * 32];
    __shared__ float csum[64], csq[64];
    const int t = threadIdx.x;
    const int b = blockIdx.y;
    const int p0 = blockIdx.x * 128;
    if (t < 64) { csum[t] = 0.f; csq[t] = 0.f; }
    const int lane = t & 31, w = t >> 5;
    const int mtile = w & 3, ngrp = w >> 2;
    const int half = lane >> 4, m = lane & 15;
    v8f acc[4] = {};
    const float* xb = x + (size_t)b * CIN * HW + p0;
    for (int kc = 0; kc < 8; ++kc) {
        #pragma unroll
        for (int i = 0; i < 16; ++i) {             // x chunk 32x128, coalesced
            int idx = i * 256 + t;
            xs[idx] = xb[(size_t)(kc * 32 + (idx >> 7)) * HW + (idx & 127)];
        }
        #pragma unroll
        for (int i = 0; i < 8; ++i) {              // A chunk 64x32
            int idx = i * 256 + t;
            as[idx] = cr_w[(idx >> 5) * CIN + kc * 32 + (idx & 31)];
        }
        if (kc < 7)                                 // prefetch next K chunk
            __builtin_prefetch(&xb[(size_t)((kc + 1) * 32 + (t >> 3)) * HW + (t & 7) * 16], 0, 1);
        __syncthreads();
        #pragma unroll
        for (int kk = 0; kk < 32; kk += 4) {
            v2f a;
            a.x = as[(mtile * 16 + m) * 32 + kk + 2 * half];
            a.y = as[(mtile * 16 + m) * 32 + kk + 2 * half + 1];
            #pragma unroll
            for (int j = 0; j < 4; ++j) {
                int n = ngrp * 64 + j * 16 + m;
                v2f bb;
                bb.x = xs[(kk + 2 * half) * 128 + n];
                bb.y = xs[(kk + 2 * half + 1) * 128 + n];
                acc[j] = wmma_f32(a, bb, acc[j]);
            }
        }
        __syncthreads();
    }
    float* cs = stats;       // cr_sum
    float* cq = stats + 64;  // cr_sq
    #pragma unroll
    for (int j = 0; j < 4; ++j) {
        #pragma unroll
        for (int r = 0; r < 8; ++r) {
            int c = mtile * 16 + r + 8 * half;
            float v = acc[j][r] + cr_b[c];
            int n = ngrp * 64 + j * 16 + m;
            y_pre[(size_t)(b * Cr + c) * HW + p0 + n] = v;
            float s1 = red16(v), s2 = red16(v * v);
            if (m == 0) { atomicAdd(&csum[c], s1); atomicAdd(&csq[c], s2); }
        }
    }
    __syncthreads();
    if (t < 64) { atomicAdd(&cs[t], csum[t]); atomicAdd(&cq[t], csq[t]); }
}

// ------------------------------------------------ generic BN finalize
__global__ void k_fin(const float* __restrict__ sum, const float* __restrict__ sq,
                      const float* __restrict__ g, const float* __restrict__ be,
                      float* __restrict__ s_out, float* __restrict__ t_out,
                      int n, float invN) {
    int i = blockIdx.x * blockDim.x + threadIdx.x;
    if (i < n) {
        float mn = sum[i] * invN;
        float var = sq[i] * invN - mn * mn;
        float s = g[i] * rsqrtf(var + 1e-5f);
        s_out[i] = s;
        t_out[i] = be[i] - mn * s;
    }
}

// ------------------------------------------------ depthwise dilated conv stats
// grid (8 row-tiles, 64 C, 8 B), block 256. Tile 16 rows x 128 cols, halo 8.
__global__ __launch_bounds__(256)
void k_dwstats(const float* __restrict__ y_pre, const float* __restrict__ dw_w,
               const float* __restrict__ dw_b, const float* __restrict__ cr_s,
               const float* __restrict__ cr_t, float* __restrict__ bn1_sum,
               float* __restrict__ bn1_sq) {
    __shared__ float tile[32 * 144];
    __shared__ float ssum[4], ssq[4];
    const int t = threadIdx.x;
    const int c = blockIdx.y, b = blockIdx.z;
    const int row0 = blockIdx.x * 16;
    if (t < 4) { ssum[t] = 0.f; ssq[t] = 0.f; }
    const float s = cr_s[c], tt = cr_t[c];
    const float* yp = y_pre + (size_t)(b * Cr + c) * HW;
    for (int idx = t; idx < 32 * 144; idx += 256) {
        int r = idx / 144, cc = idx - r * 144;
        int gy = row0 - 8 + r, gx = cc - 8;
        float v = 0.f;
        if (gy >= 0 && gy < Hh && gx >= 0 && gx < Ww) {
            float u = s * yp[gy * Ww + gx] + tt;
            v = u > 0.f ? u : 0.f;
        }
        tile[idx] = v;
    }
    __syncthreads();
    float wgt[4][9], bias[4];
    #pragma unroll
    for (int d = 0; d < 4; ++d) {
        bias[d] = dw_b[d * Cr + c];
        #pragma unroll
        for (int k = 0; k < 9; ++k) wgt[d][k] = dw_w[(d * Cr + c) * 9 + k];
    }
    float lsum[4] = {0, 0, 0, 0}, lsq[4] = {0, 0, 0, 0};
    for (int i = 0; i < 8; ++i) {
        int p = t + i * 256;
        const float* ctr = &tile[(8 + (p >> 7)) * 144 + 8 + (p & 127)];
        #pragma unroll
        for (int d = 0; d < 4; ++d) {
            int dil = 1 << d;
            float o = bias[d];
            #pragma unroll
            for (int ky = 0; ky < 3; ++ky)
                #pragma unroll
                for (int kx = 0; kx < 3; ++kx)
                    o += wgt[d][ky * 3 + kx] * ctr[(ky - 1) * dil * 144 + (kx - 1) * dil];
            lsum[d] += o; lsq[d] += o * o;
        }
    }
    #pragma unroll
    for (int d = 0; d < 4; ++d) {
        float s1 = red32(lsum[d]), s2 = red32(lsq[d]);
        if ((t & 31) == 0) { atomicAdd(&ssum[d], s1); atomicAdd(&ssq[d], s2); }
    }
    __syncthreads();
    if (t < 4) {
        atomicAdd(&bn1_sum[c * 4 + t], ssum[t]);
        atomicAdd(&bn1_sq[c * 4 + t], ssq[t]);
    }
}

// ------------------------------------------------ depthwise recompute + fuse
__global__ __launch_bounds__(256)
void k_dwfuse(const float* __restrict__ y_pre, const float* __restrict__ dw_w,
              const float* __restrict__ dw_b, const float* __restrict__ cr_s,
              const float* __restrict__ cr_t, const float* __restrict__ bn1_s,
              const float* __restrict__ bn1_t, const float* __restrict__ fuse_w,
              const float* __restrict__ fuse_b, float* __restrict__ f_pre,
              float* __restrict__ bn2_sum, float* __restrict__ bn2_sq) {
    __shared__ float tile[32 * 144];
    __shared__ float ss[2];
    const int t = threadIdx.x;
    const int c = blockIdx.y, b = blockIdx.z;
    const int row0 = blockIdx.x * 16;
    if (t < 2) ss[t] = 0.f;
    const float s = cr_s[c], tt = cr_t[c];
    const float* yp = y_pre + (size_t)(b * Cr + c) * HW;
    for (int idx = t; idx < 32 * 144; idx += 256) {
        int r = idx / 144, cc = idx - r * 144;
        int gy = row0 - 8 + r, gx = cc - 8;
        float v = 0.f;
        if (gy >= 0 && gy < Hh && gx >= 0 && gx < Ww) {
            float u = s * yp[gy * Ww + gx] + tt;
            v = u > 0.f ? u : 0.f;
        }
        tile[idx] = v;
    }
    __syncthreads();
    float wgt[4][9], bias[4], s1v[4], t1v[4], fw[4];
    #pragma unroll
    for (int d = 0; d < 4; ++d) {
        bias[d] = dw_b[d * Cr + c];
        s1v[d] = bn1_s[c * 4 + d]; t1v[d] = bn1_t[c * 4 + d];
        fw[d] = fuse_w[c * 4 + d];
        #pragma unroll
        for (int k = 0; k < 9; ++k) wgt[d][k] = dw_w[(d * Cr + c) * 9 + k];
    }
    const float fb = fuse_b[c];
    float ls = 0.f, lq = 0.f;
    float* fp_out = f_pre + (size_t)(b * Cr + c) * HW;
    for (int i = 0; i < 8; ++i) {
        int p = t + i * 256;
        int rr = p >> 7, xx = p & 127;
        const float* ctr = &tile[(8 + rr) * 144 + 8 + xx];
        float fp = fb;
        #pragma unroll
        for (int d = 0; d < 4; ++d) {
            int dil = 1 << d;
            float o = bias[d];
            #pragma unroll
            for (int ky = 0; ky < 3; ++ky)
                #pragma unroll
                for (int kx = 0; kx < 3; ++kx)
                    o += wgt[d][ky * 3 + kx] * ctr[(ky - 1) * dil * 144 + (kx - 1) * dil];
            float u = s1v[d] * o + t1v[d];
            fp += fw[d] * (u > 0.f ? u : 0.f);
        }
        fp_out[(row0 + rr) * Ww + xx] = fp;
        ls += fp; lq += fp * fp;
    }
    ls = red32(ls); lq = red32(lq);
    if ((t & 31) == 0) { atomicAdd(&ss[0], ls); atomicAdd(&ss[1], lq); }
    __syncthreads();
    if (t == 0) { atomicAdd(&bn2_sum[c], ss[0]); atomicAdd(&bn2_sq[c], ss[1]); }
}

// ------------------------------------------------ apply bn2 -> f, mb stats
__global__ __launch_bounds__(256)
void k_bn2(const float* __restrict__ f_pre, const float* __restrict__ bn2_s,
           const float* __restrict__ bn2_t, float* __restrict__ f,
           float* __restrict__ mb_sum, float* __restrict__ mb_sq) {
    __shared__ float ss[2];
    const int t = threadIdx.x;
    if (t < 2) ss[t] = 0.f;
    __syncthreads();
    size_t base = (size_t)blockIdx.x * 4096;
    int c = (int)((base >> 14) & 63);
    float s = bn2_s[c], tt = bn2_t[c];
    float ls = 0.f, lq = 0.f;
    for (int i = 0; i < 16; ++i) {
        size_t idx = base + t + i * 256;
        float u = s * f_pre[idx] + tt;
        float v = u > 0.f ? u : 0.f;
        f[idx] = v;
        ls += v; lq += v * v;
    }
    ls = red32(ls); lq = red32(lq);
    if ((t & 31) == 0) { atomicAdd(&ss[0], ls); atomicAdd(&ss[1], lq); }
    __syncthreads();
    if (t == 0) { atomicAdd(&mb_sum[c], ss[0]); atomicAdd(&mb_sq[c], ss[1]); }
}

// ------------------------------------------------ z Gram (64x64) via WMMA
// z = relu(mb(f)) computed on the fly; G = Z Z^T, zsum = row sums.
__global__ __launch_bounds__(256)
void k_gram(const float* __restrict__ f, const float* __restrict__ mb_s,
            const float* __restrict__ mb_t, float* __restrict__ gram,
            float* __restrict__ zsum) {
    __shared__ float zt[64 * 65];   // stride 65 -> conflict-free column gathers
    const int t = threadIdx.x;
    const int lane = t & 31, w = t >> 5;
    const int half = lane >> 4, m = lane & 15;
    v8f acc[2] = {};
    float zacc = 0.f;
    for (int k = 0; k < 8; ++k) {
        int ci = blockIdx.x + 256 * k;        // 2048 pixel-chunks of 64
        int b = ci >> 8;
        int p0 = (ci & 255) * 64;
        __syncthreads();
        for (int i = 0; i < 16; ++i) {
            int idx = i * 256 + t;
            int ch = idx >> 6, col = idx & 63;
            float u = mb_s[ch] * f[(size_t)(b * Cr + ch) * HW + p0 + col] + mb_t[ch];
            zt[ch * 65 + col] = u > 0.f ? u : 0.f;
        }
        __syncthreads();
        if (t < 64) {
            float srow = 0.f;
            for (int col = 0; col < 64; ++col) srow += zt[t * 65 + col];
            zacc += srow;
        }
        for (int kk = 0; kk < 64; kk += 4) {
            #pragma unroll
            for (int i = 0; i < 2; ++i) {
                int tid = w * 2 + i;
                int mt = tid >> 2, nt = tid & 3;
                v2f a, bb;
                a.x = zt[(mt * 16 + m) * 65 + kk + 2 * half];
                a.y = zt[(mt * 16 + m) * 65 + kk + 2 * half + 1];
                bb.x = zt[(nt * 16 + m) * 65 + kk + 2 * half];
                bb.y = zt[(nt * 16 + m) * 65 + kk + 2 * half + 1];
                acc[i] = wmma_f32(a, bb, acc[i]);
            }
        }
    }
    if (t < 64) atomicAdd(&zsum[t], zacc);
    #pragma unroll
    for (int i = 0; i < 2; ++i) {
        int tid = w * 2 + i;
        int mt = tid >> 2, nt = tid & 3;
        #pragma unroll
        for (int r = 0; r < 8; ++r)
            atomicAdd(&gram[(mt * 16 + r + 8 * half) * 64 + nt * 16 + m], acc[i][r]);
    }
}

// ------------------------------------------------ analytic ffbn finalize
__global__ __launch_bounds__(256)
void k_finff(const float* __restrict__ zsum, const float* __restrict__ gram,
             const float* __restrict__ ff_w, const float* __restrict__ ff_b,
             const float* __restrict__ g, const float* __restrict__ be,
             float* __restrict__ ffs, float* __restrict__ fft) {
    int o = threadIdx.x;
    const float invN = 1.f / (float)NPIX;
    float wr[64];
    for (int c = 0; c < 64; ++c) wr[c] = ff_w[o * 64 + c];
    float mz = 0.f;
    for (int c = 0; c < 64; ++c) mz += wr[c] * zsum[c];
    mz *= invN;
    float bb = ff_b[o];
    float mean = mz + bb;
    float q = 0.f;
    for (int c = 0; c < 64; ++c) {
        float acc = 0.f;
        for (int cc = 0; cc < 64; ++cc) acc += gram[c * 64 + cc] * wr[cc];
        q += wr[c] * acc;
    }
    q *= invN;
    float var = (q + 2.f * bb * mz + bb * bb) - mean * mean;
    float s = g[o] * rsqrtf(var + 1e-5f);
    ffs[o] = s;
    fft[o] = be[o] - mean * s;
}

// ------------------------------------------------ GEMM2 + fused ffbn + relu
// grid (256 pixel-blocks of 64, 8 batch). M=256 full per block; 8 waves x
// (2 Mtiles x 4 Ntiles). K=64 single LDS stage.
__global__ __launch_bounds__(256)
void k_gemm2(const float* __restrict__ f, const float* __restrict__ mb_s,
             const float* __restrict__ mb_t, const float* __restrict__ ff_w,
             const float* __restrict__ ff_b, const float* __restrict__ ffs,
             const float* __restrict__ fft, float* __restrict__ out) {
    __shared__ float zs[64 * 64];
    const int t = threadIdx.x;
    const int b = blockIdx.y;
    const int p0 = blockIdx.x * 64;
    const int lane = t & 31, w = t >> 5;
    const int half = lane >> 4, m = lane & 15;
    for (int i = 0; i < 16; ++i) {
        int idx = i * 256 + t;
        int ch = idx >> 6, col = idx & 63;
        float u = mb_s[ch] * f[(size_t)(b * Cr + ch) * HW + p0 + col] + mb_t[ch];
        zs[ch * 64 + col] = u > 0.f ? u : 0.f;
    }
    __syncthreads();
    v8f acc[2][4] = {};
    for (int kk = 0; kk < 64; kk += 4) {
        v2f bfr[4];
        #pragma unroll
        for (int j = 0; j < 4; ++j) {
            int n = j * 16 + m;
            bfr[j].x = zs[(kk + 2 * half) * 64 + n];
            bfr[j].y = zs[(kk + 2 * half + 1) * 64 + n];
        }
        #pragma unroll
        for (int i = 0; i < 2; ++i) {
            int mt = w * 2 + i;
            v2f a;
            a.x = ff_w[(mt * 16 + m) * 64 + kk + 2 * half];
            a.y = ff_w[(mt * 16 + m) * 64 + kk + 2 * half + 1];
            #pragma unroll
            for (int j = 0; j < 4; ++j) acc[i][j] = wmma_f32(a, bfr[j], acc[i][j]);
        }
    }
    #pragma unroll
    for (int i = 0; i < 2; ++i) {
        int mt = w * 2 + i;
        #pragma unroll
        for (int r = 0; r < 8; ++r) {
            int co = mt * 16 + r + 8 * half;
            float s = ffs[co], tt = fft[co], bb = ff_b[co];
            #pragma unroll
            for (int j = 0; j < 4; ++j) {
                float v = s * (acc[i][j][r] + bb) + tt;
                out[(size_t)(b * COUT + co) * HW + p0 + j * 16 + m] = v > 0.f ? v : 0.f;
            }
        }
    }
}

// ===========================================================================
extern "C" void kernel_launch(void* const* d_in, const int* in_sizes, int n_in,
                              void* d_out, int out_size, void* d_ws, size_t ws_size,
                              hipStream_t stream) {
    const float* x      = (const float*)d_in[0];
    const float* cr_w   = (const float*)d_in[1];
    const float* cr_b   = (const float*)d_in[2];
    const float* cr_g   = (const float*)d_in[3];
    const float* cr_be  = (const float*)d_in[4];
    const float* dw_w   = (const float*)d_in[5];
    const float* dw_b   = (const float*)d_in[6];
    const float* bn1_g  = (const float*)d_in[7];
    const float* bn1_b  = (const float*)d_in[8];
    const float* fuse_w = (const float*)d_in[9];
    const float* fuse_b = (const float*)d_in[10];
    const float* bn2_g  = (const float*)d_in[11];
    const float* bn2_b  = (const float*)d_in[12];
    const float* mb_g   = (const float*)d_in[13];
    const float* mb_b   = (const float*)d_in[14];
    const float* ff_w   = (const float*)d_in[15];
    const float* ff_b   = (const float*)d_in[16];
    const float* ffbn_g = (const float*)d_in[17];
    const float* ffbn_b = (const float*)d_in[18];
    float* out = (float*)d_out;
    float* ws  = (float*)d_ws;

    // stats layout inside first 16384 floats of ws
    float* cr_sum  = ws + 0;    float* cr_sq  = ws + 64;
    float* bn1_sum = ws + 128;  float* bn1_sq = ws + 384;
    float* bn2_sum = ws + 640;  float* bn2_sq = ws + 704;
    float* mb_sum  = ws + 768;  float* mb_sq  = ws + 832;
    float* zsum    = ws + 896;
    float* gram    = ws + 1024;
    float* cr_s  = ws + 5120;   float* cr_t  = ws + 5184;
    float* bn1_s = ws + 5248;   float* bn1_t = ws + 5504;
    float* bn2_s = ws + 5760;   float* bn2_t = ws + 5824;
    float* mb_s  = ws + 5888;   float* mb_t  = ws + 5952;
    float* ffs   = ws + 6016;   float* fft   = ws + 6272;
    float* bufA = ws + 16384;                  // y_pre, later reused as f
    float* bufB = bufA + (size_t)Bc * Cr * HW; // f_pre

    const float invN = 1.f / (float)NPIX;

    k_init<<<64, 256, 0, stream>>>(ws);
    k_gemm1<<<dim3(128, 8), 256, 0, stream>>>(x, cr_w, cr_b, bufA, ws);
    k_fin<<<1, 64, 0, stream>>>(cr_sum, cr_sq, cr_g, cr_be, cr_s, cr_t, 64, invN);
    k_dwstats<<<dim3(8, 64, 8), 256, 0, stream>>>(bufA, dw_w, dw_b, cr_s, cr_t,
                                                  bn1_sum, bn1_sq);
    k_fin<<<1, 256, 0, stream>>>(bn1_sum, bn1_sq, bn1_g, bn1_b, bn1_s, bn1_t, 256, invN);
    k_dwfuse<<<dim3(8, 64, 8), 256, 0, stream>>>(bufA, dw_w, dw_b, cr_s, cr_t,
                                                 bn1_s, bn1_t, fuse_w, fuse_b,
                                                 bufB, bn2_sum, bn2_sq);
    k_fin<<<1, 64, 0, stream>>>(bn2_sum, bn2_sq, bn2_g, bn2_b, bn2_s, bn2_t, 64, invN);
    k_bn2<<<2048, 256, 0, stream>>>(bufB, bn2_s, bn2_t, bufA, mb_sum, mb_sq);
    k_fin<<<1, 64, 0, stream>>>(mb_sum, mb_sq, mb_g, mb_b, mb_s, mb_t, 64, invN);
    k_gram<<<256, 256, 0, stream>>>(bufA, mb_s, mb_t, gram, zsum);
    k_finff<<<1, 256, 0, stream>>>(zsum, gram, ff_w, ff_b, ffbn_g, ffbn_b, ffs, fft);
    k_gemm2<<<dim3(256, 8), 256, 0, stream>>>(bufA, mb_s, mb_t, ff_w, ff_b,
                                              ffs, fft, out);
}